// HierarchicalConsensus_60301340836403
// MI455X (gfx1250) — compile-verified
//
#include <hip/hip_runtime.h>
#include <math.h>

// ---------------- problem constants ----------------
#define B_SZ    8
#define N_AG    1024
#define NL_SZ   64
#define D_DIM   256
#define MSG_DIM 256
#define K_AG    16
#define KL_SZ   8
#define NPB     8        // nodes per block (1 node per wave, 8 waves)

// ---------------- LDS layout (floats) ----------------
// A/H tiles use row stride 260 (TDM pad: 256 DW interval + 4 DW pad) -> conflict-free
// W slice uses row stride 40 (TDM pad: 32 DW interval + 8 DW pad)    -> conflict-free
#define A_STR    260
#define NODE_STR (16 * A_STR)            // 4160 floats / node
#define CAT_STR  516
#define W_STR    40

#define A_OFF    0                       // NPB*NODE_STR = 33280   (cat rows alias this)
#define H_OFF    (NPB * NODE_STR)        // 33280 floats          (upd rows alias this)
#define W_OFF    (2 * NPB * NODE_STR)    // 256*W_STR = 10240 floats
#define WT_OFF   (W_OFF + 256 * W_STR)   // NPB*16 = 128 floats
#define LDS_FLOATS (WT_OFF + NPB * 16)   // 76928 floats
#define LDS_BYTES  (LDS_FLOATS * 4)      // 307712 bytes (<= 320KB per WGP)

typedef __attribute__((ext_vector_type(2))) float        v2f;
typedef __attribute__((ext_vector_type(8))) float        v8f;
typedef __attribute__((ext_vector_type(4))) unsigned int v4u;
typedef __attribute__((ext_vector_type(8))) int          v8i;
typedef __attribute__((ext_vector_type(4))) int          v4i;

__device__ __forceinline__ float gelu_exact(float x) {
    return 0.5f * x * (1.0f + erff(x * 0.70710678118654752440f));
}

// ---- Tensor Data Mover: build D# per CDNA5 ISA ch.8 and issue tensor_load_to_lds ----
// clang-23 form: (v4u group0, v8i group1, v4i group2, v4i group3, v8i extra, i32 cpol)
// g1flags packs: workgroup_mask=0 | data_size<<16 | pad_enable<<20 | pad_interval<<22 | pad_amount<<25
__device__ __forceinline__ void tdm_load(unsigned g0flags, unsigned lds_byte,
                                         const void* gptr, unsigned g1flags,
                                         unsigned td0, unsigned td1,
                                         unsigned tile0, unsigned tile1,
                                         unsigned stride0, v4i g2, v4i g3)
{
    unsigned long long ga = (unsigned long long)(uintptr_t)gptr;
    v4u g0;
    g0[0] = g0flags;                                   // count=1 (+gather bits)
    g0[1] = lds_byte;                                  // lds_addr
    g0[2] = (unsigned)(ga & 0xffffffffu);              // global_addr[31:0]
    g0[3] = (unsigned)((ga >> 32) & 0x01ffffffu)       // global_addr[56:32]
          | (2u << 30);                                // type = 2 ("image")
    v8i g1;
    g1[0] = (int)g1flags;
    g1[1] = (int)((td0 & 0xffffu) << 16);              // atomic_barrier=0 | tensor_dim0 lo
    g1[2] = (int)((td0 >> 16) | ((td1 & 0xffffu) << 16));
    g1[3] = (int)((td1 >> 16) | (tile0 << 16));        // tensor_dim1 hi | tile_dim0
    g1[4] = (int)(tile1 & 0xffffu);                    // tile_dim1 | tile_dim2=0
    g1[5] = (int)stride0;                              // tensor_dim0_stride[31:0]
    g1[6] = 0;
    g1[7] = 0;
    v8i gx = { 0, 0, 0, 0, 0, 0, 0, 0 };
    __builtin_amdgcn_tensor_load_to_lds(g0, g1, g2, g3, gx, 0);
}

// data_size=4B, pad: interval 256 DW (enc 7), amount 4 DW (enc 3) -> LDS row stride 260 floats
#define TDM_FLAGS_GATHER 0x07D20000u
// data_size=4B, pad: interval 32 DW (enc 4), amount 8 DW (enc 7)  -> LDS row stride 40 floats
#define TDM_FLAGS_WSLICE 0x0F120000u

// One fused message-passing step.  feat:[B,Nn,D]  nbr_idx:[B,Nn,K]  nbr_w:[B,Nn,K]  out:[B,Nn,D]
__global__ void __launch_bounds__(256, 1)
mp_step_kernel(const float* __restrict__ feat,
               const int*   __restrict__ nbr_idx,
               const float* __restrict__ nbr_w,
               const float* __restrict__ w1, const float* __restrict__ b1,
               const float* __restrict__ w2, const float* __restrict__ b2,
               const float* __restrict__ wu, const float* __restrict__ bu,
               const float* __restrict__ lng, const float* __restrict__ lnb,
               float* __restrict__ out,
               int Nn, int K, int useMask)
{
    extern __shared__ float lds[];
    const int lane  = threadIdx.x & 31;
    const int wave  = threadIdx.x >> 5;          // local node 0..7
    const int gnode = blockIdx.x * NPB + wave;   // flat (b*Nn + n)
    const int bb    = gnode / Nn;
    const int mrow  = lane & 15;                 // WMMA M (A) / N (B,C) index
    const int khalf = (lane >> 4) << 1;          // lanes 0-15 -> k{0,1}, 16-31 -> k{2,3}
    const int hi8   = (lane >> 4) << 3;          // C-layout row offset

    // ================= Phase 0: TDM gather of neighbor rows + weights =================
    const long nbrBase = (long)gnode * K;
    {
        // pack 16 row indices (16-bit, wave-uniform) for gather-mode TDM
        unsigned p[8];
        #pragma unroll
        for (int j = 0; j < 8; ++j) {
            int i0 = (2 * j     < K) ? nbr_idx[nbrBase + 2 * j]     : 0;
            int i1 = (2 * j + 1 < K) ? nbr_idx[nbrBase + 2 * j + 1] : 0;
            if (i0 < 0) i0 = 0;
            if (i1 < 0) i1 = 0;
            p[j] = (unsigned)i0 | ((unsigned)i1 << 16);
        }
        v4i g2 = { (int)p[0], (int)p[1], (int)p[2], (int)p[3] };
        v4i g3 = { (int)p[4], (int)p[5], (int)p[6], (int)p[7] };
        // gather 16 rows x 256 floats from this batch's feature table into Abuf[wave]
        tdm_load(0x80000001u /*count=1 | gather_mode*/,
                 (unsigned)((A_OFF + wave * NODE_STR) * 4),
                 feat + (long)bb * Nn * D_DIM,
                 TDM_FLAGS_GATHER,
                 /*td0*/ D_DIM, /*td1*/ (unsigned)Nn,
                 /*tile0*/ D_DIM, /*tile1(=#idx)*/ 16,
                 /*stride0*/ D_DIM, g2, g3);
    }
    // neighbor weights (masked)
    if (lane < 16) {
        float w = 0.f;
        if (lane < K) {
            w = nbr_w[nbrBase + lane];
            if (useMask) {
                int idx = nbr_idx[nbrBase + lane];
                if (idx < 0) w = 0.f;
            }
        }
        lds[WT_OFF + wave * 16 + lane] = w;
    }
    __builtin_amdgcn_s_wait_tensorcnt(0);   // own gather complete (wave-private A tile)

    v4i zz = { 0, 0, 0, 0 };

    // ================= Phase 1: H = GELU(A @ w1 + b1), 2 column tiles / iter ============
    for (int it = 0; it < 8; ++it) {
        const int n0 = it * 32;
        if (wave == 0) {
            // stage w1[:, n0:n0+32] row-major into Wslice (stride 40)
            tdm_load(1u, (unsigned)(W_OFF * 4), w1 + n0, TDM_FLAGS_WSLICE,
                     32, MSG_DIM, 32, MSG_DIM, MSG_DIM, zz, zz);
            __builtin_amdgcn_s_wait_tensorcnt(0);
        }
        __syncthreads();

        v8f c0 = {}, c1 = {};
        const float* Ap = &lds[A_OFF + wave * NODE_STR + mrow * A_STR + khalf];
        #pragma unroll 4
        for (int kk = 0; kk < 64; ++kk) {
            v2f a = *(const v2f*)(Ap + kk * 4);
            const int d = kk * 4 + khalf;
            v2f b0, b1v;
            b0.x  = lds[W_OFF + d * W_STR + mrow];
            b0.y  = lds[W_OFF + (d + 1) * W_STR + mrow];
            b1v.x = lds[W_OFF + d * W_STR + 16 + mrow];
            b1v.y = lds[W_OFF + (d + 1) * W_STR + 16 + mrow];
            c0 = __builtin_amdgcn_wmma_f32_16x16x4_f32(false, a, false, b0,
                                                       (short)0, c0, false, false);
            c1 = __builtin_amdgcn_wmma_f32_16x16x4_f32(false, a, false, b1v,
                                                       (short)0, c1, false, false);
        }
        const float bias0 = b1[n0 + mrow];
        const float bias1 = b1[n0 + 16 + mrow];
        float* Hn = &lds[H_OFF + wave * NODE_STR];
        #pragma unroll
        for (int r = 0; r < 8; ++r) {
            const int M = r + hi8;
            Hn[M * A_STR + n0 + mrow]      = gelu_exact(c0[r] + bias0);
            Hn[M * A_STR + n0 + 16 + mrow] = gelu_exact(c1[r] + bias1);
        }
        __syncthreads();
    }

    // ===== Build cat rows (aliases dead A region): [own feat | agg], zero pad rows 8..15
    {
        const float* src = feat + (long)gnode * D_DIM;
        #pragma unroll
        for (int j = 0; j < 8; ++j)
            lds[A_OFF + wave * CAT_STR + lane + j * 32] = src[lane + j * 32];
        #pragma unroll
        for (int j = 0; j < 16; ++j)
            lds[A_OFF + (wave + 8) * CAT_STR + lane + j * 32] = 0.f;
    }

    // ================= Phase 2: msgs = H @ w2 + b2 ; agg = sum_k wt[k]*msgs[k] ==========
    for (int it = 0; it < 8; ++it) {
        const int n0 = it * 32;
        if (wave == 0) {
            tdm_load(1u, (unsigned)(W_OFF * 4), w2 + n0, TDM_FLAGS_WSLICE,
                     32, MSG_DIM, 32, MSG_DIM, MSG_DIM, zz, zz);
            __builtin_amdgcn_s_wait_tensorcnt(0);
        }
        __syncthreads();

        v8f c0 = {}, c1 = {};
        const float* Ap = &lds[H_OFF + wave * NODE_STR + mrow * A_STR + khalf];
        #pragma unroll 4
        for (int kk = 0; kk < 64; ++kk) {
            v2f a = *(const v2f*)(Ap + kk * 4);
            const int d = kk * 4 + khalf;
            v2f b0, b1v;
            b0.x  = lds[W_OFF + d * W_STR + mrow];
            b0.y  = lds[W_OFF + (d + 1) * W_STR + mrow];
            b1v.x = lds[W_OFF + d * W_STR + 16 + mrow];
            b1v.y = lds[W_OFF + (d + 1) * W_STR + 16 + mrow];
            c0 = __builtin_amdgcn_wmma_f32_16x16x4_f32(false, a, false, b0,
                                                       (short)0, c0, false, false);
            c1 = __builtin_amdgcn_wmma_f32_16x16x4_f32(false, a, false, b1v,
                                                       (short)0, c1, false, false);
        }
        const float bias0 = b2[n0 + mrow];
        const float bias1 = b2[n0 + 16 + mrow];
        // weighted reduce over neighbor (M) dim: c[r] sits at M = r + hi8, col = n0(+16) + mrow
        float s0 = 0.f, s1 = 0.f;
        #pragma unroll
        for (int r = 0; r < 8; ++r) {
            const float wt = lds[WT_OFF + wave * 16 + r + hi8];
            s0 += (c0[r] + bias0) * wt;
            s1 += (c1[r] + bias1) * wt;
        }
        s0 += __shfl_xor(s0, 16, 32);
        s1 += __shfl_xor(s1, 16, 32);
        if (lane < 16) {
            lds[A_OFF + wave * CAT_STR + 256 + n0 + lane]      = s0;
            lds[A_OFF + wave * CAT_STR + 256 + n0 + 16 + lane] = s1;
        }
        __syncthreads();
    }

    // ================= Phase 3: upd = cat @ wu + bu  (M=16 incl. zero pad rows) =========
    // Each wave owns 2 of the 16 column tiles; wu B fragments streamed from L2,
    // one shared A-fragment stream feeds both accumulators.
    {
        const int n0a = wave * 16;
        const int n0b = (wave + 8) * 16;
        v8f c0 = {}, c1 = {};
        const float* Ap = &lds[A_OFF + mrow * CAT_STR + khalf];
        #pragma unroll 4
        for (int kk = 0; kk < 128; ++kk) {
            v2f a = *(const v2f*)(Ap + kk * 4);
            const int k0 = kk * 4 + khalf;
            v2f b0, b1v;
            b0.x  = wu[(long)k0 * D_DIM + n0a + mrow];
            b0.y  = wu[(long)(k0 + 1) * D_DIM + n0a + mrow];
            b1v.x = wu[(long)k0 * D_DIM + n0b + mrow];
            b1v.y = wu[(long)(k0 + 1) * D_DIM + n0b + mrow];
            c0 = __builtin_amdgcn_wmma_f32_16x16x4_f32(false, a, false, b0,
                                                       (short)0, c0, false, false);
            c1 = __builtin_amdgcn_wmma_f32_16x16x4_f32(false, a, false, b1v,
                                                       (short)0, c1, false, false);
        }
        const float bias0 = bu[n0a + mrow];
        const float bias1 = bu[n0b + mrow];
        #pragma unroll
        for (int r = 0; r < 8; ++r) {
            const int M = r + hi8;
            if (M < 8) {   // rows 8..15 are padding
                lds[H_OFF + M * A_STR + n0a + mrow] = c0[r] + bias0;  // upd aliases H
                lds[H_OFF + M * A_STR + n0b + mrow] = c1[r] + bias1;
            }
        }
    }
    __syncthreads();

    // ================= Phase 4: out = GELU(LayerNorm(upd)) =================
    {
        float xv[8], ssum = 0.f, ssq = 0.f;
        #pragma unroll
        for (int j = 0; j < 8; ++j) {
            xv[j] = lds[H_OFF + wave * A_STR + lane + j * 32];
            ssum += xv[j];
            ssq  += xv[j] * xv[j];
        }
        #pragma unroll
        for (int off = 16; off > 0; off >>= 1) {
            ssum += __shfl_xor(ssum, off, 32);
            ssq  += __shfl_xor(ssq,  off, 32);
        }
        const float mu   = ssum * (1.0f / 256.0f);
        const float var  = ssq * (1.0f / 256.0f) - mu * mu;
        const float rstd = rsqrtf(var + 1e-5f);
        float* dst = out + (long)gnode * D_DIM;
        #pragma unroll
        for (int j = 0; j < 8; ++j) {
            const int col = lane + j * 32;
            const float y = (xv[j] - mu) * rstd * lng[col] + lnb[col];
            dst[col] = gelu_exact(y);
        }
    }
}

extern "C" void kernel_launch(void* const* d_in, const int* in_sizes, int n_in,
                              void* d_out, int out_size, void* d_ws, size_t ws_size,
                              hipStream_t stream) {
    const float* agent_feat  = (const float*)d_in[0];
    const int*   nidx        = (const int*)  d_in[1];
    const float* nw          = (const float*)d_in[2];
    const float* leader_feat = (const float*)d_in[3];
    const int*   lidx        = (const int*)  d_in[4];
    const float* lw          = (const float*)d_in[5];

    // Canonical order: w1,b1,w2,b2,wu,bu,ln_g,ln_b
    const float* intra[8];
    const float* inter[8];
    if (in_sizes[6] == D_DIM * MSG_DIM) {
        // dict insertion order: params['intra'] then params['inter']
        for (int i = 0; i < 8; ++i) {
            intra[i] = (const float*)d_in[6 + i];
            inter[i] = (const float*)d_in[14 + i];
        }
    } else {
        // jax pytree sorted keys: 'inter' before 'intra'; inner b1,b2,bu,ln_b,ln_g,w1,w2,wu
        const int remap[8] = {5, 0, 6, 1, 7, 2, 4, 3};
        for (int i = 0; i < 8; ++i) {
            inter[i] = (const float*)d_in[6 + remap[i]];
            intra[i] = (const float*)d_in[14 + remap[i]];
        }
    }

    float* out_agents  = (float*)d_out;                                  // B*N*D
    float* out_leaders = (float*)d_out + (size_t)B_SZ * N_AG * D_DIM;    // B*NL*D
    float* tmp_a       = (float*)d_ws;                                   // B*N*D
    float* tmp_l       = (float*)d_ws + (size_t)B_SZ * N_AG * D_DIM;     // B*NL*D

    (void)hipFuncSetAttribute(reinterpret_cast<const void*>(&mp_step_kernel),
                              hipFuncAttributeMaxDynamicSharedMemorySize, LDS_BYTES);

    dim3 blk(256, 1, 1);
    dim3 g_intra((B_SZ * N_AG) / NPB, 1, 1);   // 1024 blocks
    dim3 g_inter((B_SZ * NL_SZ) / NPB, 1, 1);  // 64 blocks

    // Intra: 2 steps, ping-pong via workspace
    mp_step_kernel<<<g_intra, blk, LDS_BYTES, stream>>>(
        agent_feat, nidx, nw,
        intra[0], intra[1], intra[2], intra[3], intra[4], intra[5], intra[6], intra[7],
        tmp_a, N_AG, K_AG, 0);
    mp_step_kernel<<<g_intra, blk, LDS_BYTES, stream>>>(
        tmp_a, nidx, nw,
        intra[0], intra[1], intra[2], intra[3], intra[4], intra[5], intra[6], intra[7],
        out_agents, N_AG, K_AG, 0);

    // Inter (leaders): 2 steps with -1 mask
    mp_step_kernel<<<g_inter, blk, LDS_BYTES, stream>>>(
        leader_feat, lidx, lw,
        inter[0], inter[1], inter[2], inter[3], inter[4], inter[5], inter[6], inter[7],
        tmp_l, NL_SZ, KL_SZ, 1);
    mp_step_kernel<<<g_inter, blk, LDS_BYTES, stream>>>(
        tmp_l, lidx, lw,
        inter[0], inter[1], inter[2], inter[3], inter[4], inter[5], inter[6], inter[7],
        out_leaders, NL_SZ, KL_SZ, 1);
}